// CTC_43044162241024
// MI455X (gfx1250) — compile-verified
//
#include <hip/hip_runtime.h>
#include <hip/hip_bf16.h>
#include <math.h>

#define Nn 16
#define Tt 1024
#define KDIM 512
#define Vv 4096
#define Ss 128
#define Ll 257              // 2*S+1
#define Mrows (Nn * Tt)     // 16384
#define CB 32               // number of 128-wide column blocks over V
#define NEGV (-1e30f)

typedef __attribute__((ext_vector_type(2))) float v2f;
typedef __attribute__((ext_vector_type(8))) float v8f;

__device__ __forceinline__ float logaddexpf_(float a, float b) {
    float m = fmaxf(a, b);
    float d = fminf(a, b) - m;
    return m + log1pf(__expf(d));
}

// ---------------------------------------------------------------------------
// Kernel 1: fp32 WMMA GEMM (logits tile) fused with streaming logsumexp
// partials.  Block = 256 threads (8 waves).  Each block computes a 16-row x
// 128-col logits tile; each wave owns one 16x16 WMMA tile and runs the full
// K=512 contraction as 128 x v_wmma_f32_16x16x4_f32.
// A (hs) tile is DMA'd global->LDS with the CDNA5 async path (no VGPR
// bounce), padded to stride 516 so the 16-lane row reads are bank-conflict
// free.  Epilogue: per-row max/sumexp over the 128 columns, written as
// (max, sum) partials per (row, colblock).
// ---------------------------------------------------------------------------
__global__ __launch_bounds__(256)
void gemm_lse_partial(const float* __restrict__ hs,
                      const float* __restrict__ W,
                      const float* __restrict__ bias,
                      float* __restrict__ pmax,
                      float* __restrict__ psum) {
    __shared__ float Alds[16 * 516];
    __shared__ float wmaxs[8][16];
    __shared__ float wsums[8][16];

    const int cb   = blockIdx.x;       // 0..31 : 128-col block of V
    const int mt   = blockIdx.y;       // 0..1023 : 16-row tile of M
    const int tid  = threadIdx.x;
    const int wave = tid >> 5;
    const int lane = tid & 31;
    const int half = lane >> 4;        // 0: K pair (k,k+1), 1: (k+2,k+3)
    const int lr   = lane & 15;        // row (A) / col (B) index in tile

    // Async global->LDS staging of the A tile.  Rows mt*16 .. mt*16+15 are
    // contiguous in hs ((N,T,K) row-major) = one 32KB chunk; each thread
    // issues 8 x b128 async copies with per-lane padded LDS destinations.
    const float* abase = hs + (size_t)mt * (16 * KDIM);
    for (int i = tid; i < 2048; i += 256) {          // 2048 float4 = 8192 f
        const int e = i * 4;
        const int r = e >> 9;          // / 512
        const int c = e & 511;
        const unsigned lds_addr =
            (unsigned)(uintptr_t)(&Alds[r * 516 + c]);     // low 32b = LDS byte addr
        const unsigned long long gaddr =
            (unsigned long long)(uintptr_t)(abase + e);
        asm volatile("global_load_async_to_lds_b128 %0, %1, off"
                     :: "v"(lds_addr), "v"(gaddr)
                     : "memory");
    }
    asm volatile("s_wait_asynccnt 0x0" ::: "memory");
    __syncthreads();

    const int v0 = cb * 128 + wave * 16;             // this wave's 16 columns
    const float* wrow = W + (size_t)(v0 + lr) * KDIM; // B column = W row
    const float* arow = &Alds[lr * 516];

    v8f acc = {};
    #pragma unroll 8
    for (int k0 = 0; k0 < KDIM; k0 += 4) {
        const int ka = k0 + 2 * half;
        const float2 av = *reinterpret_cast<const float2*>(&arow[ka]); // ds b64
        const float2 bv = *reinterpret_cast<const float2*>(&wrow[ka]); // global b64
        v2f afrag; afrag.x = av.x; afrag.y = av.y;
        v2f bfrag; bfrag.x = bv.x; bfrag.y = bv.y;
        // D = A(16x4, fp32) x B(4x16, fp32) + C(16x16, fp32)
        acc = __builtin_amdgcn_wmma_f32_16x16x4_f32(
            false, afrag, false, bfrag, (short)0, acc, false, false);
    }

    const float bval = bias[v0 + lr];  // column bias (same col for all 8 rows)

    // Per-row reduction over this wave's 16 columns.
    // VGPR j: lanes 0-15 -> row j, lanes 16-31 -> row j+8 (cols = lr).
    #pragma unroll
    for (int j = 0; j < 8; ++j) {
        float x = acc[j] + bval;
        float m = x;
        m = fmaxf(m, __shfl_xor(m, 8, 32));
        m = fmaxf(m, __shfl_xor(m, 4, 32));
        m = fmaxf(m, __shfl_xor(m, 2, 32));
        m = fmaxf(m, __shfl_xor(m, 1, 32));
        float s = __expf(x - m);
        s += __shfl_xor(s, 8, 32);
        s += __shfl_xor(s, 4, 32);
        s += __shfl_xor(s, 2, 32);
        s += __shfl_xor(s, 1, 32);
        if (lr == 0) {
            const int r = j + 8 * half;
            wmaxs[wave][r] = m;
            wsums[wave][r] = s;
        }
    }
    __syncthreads();

    // Combine the 8 waves' partials -> one (max,sum) per row for this block.
    if (tid < 16) {
        float gm = NEGV;
        #pragma unroll
        for (int w = 0; w < 8; ++w) gm = fmaxf(gm, wmaxs[w][tid]);
        float gs = 0.f;
        #pragma unroll
        for (int w = 0; w < 8; ++w) gs += wsums[w][tid] * __expf(wmaxs[w][tid] - gm);
        const int m = mt * 16 + tid;
        pmax[(size_t)m * CB + cb] = gm;
        psum[(size_t)m * CB + cb] = gs;
    }
}

// ---------------------------------------------------------------------------
// Kernel 2: combine the 32 column-block partials into lse[m].
// ---------------------------------------------------------------------------
__global__ void lse_combine(const float* __restrict__ pmax,
                            const float* __restrict__ psum,
                            float* __restrict__ lse) {
    const int m = blockIdx.x * blockDim.x + threadIdx.x;
    if (m >= Mrows) return;
    float gm = NEGV;
    for (int i = 0; i < CB; ++i) gm = fmaxf(gm, pmax[(size_t)m * CB + i]);
    float gs = 0.f;
    for (int i = 0; i < CB; ++i)
        gs += psum[(size_t)m * CB + i] * __expf(pmax[(size_t)m * CB + i] - gm);
    lse[m] = gm + logf(gs);
}

// ---------------------------------------------------------------------------
// Kernel 3: log-probs at the 129 vocab entries CTC actually reads
// (blank=0 plus the S labels of each sample).  Block per (n,t) row; hs row
// staged in LDS; one wave per dot product, lane-strided MACs + shuffle tree.
// ---------------------------------------------------------------------------
__global__ __launch_bounds__(256)
void label_logits(const float* __restrict__ hs,
                  const float* __restrict__ W,
                  const float* __restrict__ bias,
                  const int* __restrict__ ys,
                  const float* __restrict__ lse,
                  float* __restrict__ lpb,
                  float* __restrict__ lpl) {
    __shared__ float hrow[KDIM];
    const int m    = blockIdx.x;        // n*T + t
    const int n    = m >> 10;
    const int tid  = threadIdx.x;
    const int wave = tid >> 5;
    const int lane = tid & 31;

    for (int i = tid; i < KDIM; i += 256) hrow[i] = hs[(size_t)m * KDIM + i];
    __syncthreads();

    const float l = lse[m];
    for (int j = wave; j < 129; j += 8) {
        const int v = (j == 0) ? 0 : ys[n * Ss + (j - 1)];
        const float* wr = W + (size_t)v * KDIM;
        float s = 0.f;
        #pragma unroll 4
        for (int e = lane; e < KDIM; e += 32) s = fmaf(hrow[e], wr[e], s);
        s += __shfl_xor(s, 16, 32);
        s += __shfl_xor(s,  8, 32);
        s += __shfl_xor(s,  4, 32);
        s += __shfl_xor(s,  2, 32);
        s += __shfl_xor(s,  1, 32);
        if (lane == 0) {
            const float r = s + bias[v] - l;
            if (j == 0) lpb[m] = r;
            else        lpl[(size_t)m * Ss + (j - 1)] = r;
        }
    }
}

// ---------------------------------------------------------------------------
// Kernel 4: CTC alpha recursion.  One block per sample; thread s owns
// alpha[s] (s < 257); double-buffered LDS with 2-slot NEG padding so the
// s-1 / s-2 reads need no branches; one barrier per timestep.
// ---------------------------------------------------------------------------
__global__ __launch_bounds__(288)
void ctc_forward(const int* __restrict__ hlens,
                 const int* __restrict__ ys,
                 const int* __restrict__ ylens,
                 const float* __restrict__ lpb,
                 const float* __restrict__ lpl,
                 float* __restrict__ loss) {
    __shared__ float bufA[Ll + 2];
    __shared__ float bufB[Ll + 2];
    const int n = blockIdx.x;
    const int s = threadIdx.x;
    const int hlen = hlens[n];
    const int Ln   = 2 * ylens[n] + 1;
    const bool active = (s < Ll);

    int ext = 0, extp2 = 0;
    if (active && (s & 1)) ext = ys[n * Ss + (s >> 1)];
    if (s >= 3 && (s & 1)) extp2 = ys[n * Ss + ((s - 2) >> 1)];
    const bool allow = (s >= 2) && (ext != 0) && (ext != extp2);

    // t = 0 init: only s==0 (blank) and s==1 (first label) are live.
    if (active) {
        float a0 = NEGV;
        if (s == 0) a0 = lpb[(size_t)n * Tt];
        if (s == 1) a0 = lpl[(size_t)n * Tt * Ss];
        bufA[2 + s] = a0;
    }
    if (s == 0) { bufA[0] = NEGV; bufA[1] = NEGV; bufB[0] = NEGV; bufB[1] = NEGV; }
    __syncthreads();

    float* cur = bufA;
    float* nxt = bufB;
    for (int t = 1; t < Tt; ++t) {
        if (active) {
            const float a1 = cur[2 + s];
            const float a2 = cur[1 + s];
            const float a3 = allow ? cur[s] : NEGV;
            const float lp = (s & 1)
                ? lpl[((size_t)n * Tt + t) * Ss + (s >> 1)]
                : lpb[(size_t)n * Tt + t];
            float v = logaddexpf_(logaddexpf_(a1, a2), a3) + lp;
            if (s >= Ln) v = NEGV;
            nxt[2 + s] = (t < hlen) ? v : a1;   // freeze past hlen
        }
        float* tmp = cur; cur = nxt; nxt = tmp;
        __syncthreads();
    }

    if (s == 0) {
        const float ll = logaddexpf_(cur[2 + Ln - 1], cur[2 + Ln - 2]);
        loss[n] = (ll > -1e29f) ? -ll : 0.f;    // zero_infinity
    }
}

// ---------------------------------------------------------------------------
// Kernel 5: deterministic ordered reduction of the 16 sample losses.
// ---------------------------------------------------------------------------
__global__ void finalize(const float* __restrict__ loss, float* __restrict__ out) {
    if (blockIdx.x == 0 && threadIdx.x == 0) {
        float s = 0.f;
        for (int i = 0; i < Nn; ++i) s += loss[i];
        out[0] = s / (float)Nn;
    }
}

extern "C" void kernel_launch(void* const* d_in, const int* in_sizes, int n_in,
                              void* d_out, int out_size, void* d_ws, size_t ws_size,
                              hipStream_t stream) {
    (void)in_sizes; (void)n_in; (void)out_size; (void)ws_size;
    const float* hs    = (const float*)d_in[0];
    const int*   hlens = (const int*)d_in[1];
    const int*   ys    = (const int*)d_in[2];
    const int*   ylens = (const int*)d_in[3];
    const float* W     = (const float*)d_in[4];
    const float* bias  = (const float*)d_in[5];
    float* out = (float*)d_out;

    // Workspace layout (floats), ~12.7 MB total:
    float* ws   = (float*)d_ws;
    float* pmax = ws;                                  // Mrows*CB
    float* psum = pmax + (size_t)Mrows * CB;           // Mrows*CB
    float* lse  = psum + (size_t)Mrows * CB;           // Mrows
    float* lpb  = lse  + Mrows;                        // Mrows
    float* lpl  = lpb  + Mrows;                        // Mrows*Ss
    float* loss = lpl  + (size_t)Mrows * Ss;           // Nn

    gemm_lse_partial<<<dim3(CB, Mrows / 16), 256, 0, stream>>>(hs, W, bias, pmax, psum);
    lse_combine<<<(Mrows + 255) / 256, 256, 0, stream>>>(pmax, psum, lse);
    label_logits<<<Mrows, 256, 0, stream>>>(hs, W, bias, ys, lse, lpb, lpl);
    ctc_forward<<<Nn, 288, 0, stream>>>(hlens, ys, ylens, lpb, lpl, loss);
    finalize<<<1, 32, 0, stream>>>(loss, out);
}